// GDNLayer_42116449305306
// MI455X (gfx1250) — compile-verified
//
#include <hip/hip_runtime.h>

#define IN_DIM  128
#define OUT_DIM 64

typedef __attribute__((ext_vector_type(2))) float v2f;
typedef __attribute__((ext_vector_type(8))) float v8f;

// ---- monotone order-preserving float<->uint encoding for atomic max ----
__device__ __forceinline__ unsigned f2key(float f) {
    unsigned b = __float_as_uint(f);
    return (b & 0x80000000u) ? ~b : (b | 0x80000000u);
}
__device__ __forceinline__ float key2f(unsigned k) {
    unsigned b = (k & 0x80000000u) ? (k & 0x7FFFFFFFu) : ~k;
    return __uint_as_float(b);
}

// ---- Kernel 1: z = h@W1^T and h_d = h@W2^T via f32 WMMA (16x16x4) ----
// One wave per 16-row node tile; computes all 4 column tiles for both W1 and W2,
// reusing the A (h) fragment. Per k-step: batch all 9 independent loads first,
// then issue 8 back-to-back WMMAs so load latency overlaps matrix issue.
__global__ void __launch_bounds__(128)
k_gemm(const float* __restrict__ h, const float* __restrict__ W1,
       const float* __restrict__ W2, float* __restrict__ z,
       float* __restrict__ hd, int n) {
    const int lane = threadIdx.x & 31;
    const int wave = (int)((blockIdx.x * blockDim.x + threadIdx.x) >> 5);
    const int ntiles = (n + 15) >> 4;
    if (wave >= ntiles) return;               // wave-uniform: EXEC stays all-1s

    const int row0 = wave * 16;
    const int m  = lane & 15;                 // A: row-in-tile, B: col-in-tile
    const int kq = (lane >> 4) * 2;           // K sub-offset within 4-wide K block

    int arow = row0 + m; if (arow >= n) arow = n - 1;   // clamp loads, guard stores
    const float* hrow = h + (size_t)arow * IN_DIM + kq;
    const float* w1p  = W1 + (size_t)m * IN_DIM + kq;   // + t*16*IN_DIM per col tile
    const float* w2p  = W2 + (size_t)m * IN_DIM + kq;

    v8f accz[4] = {}; v8f accd[4] = {};

#pragma unroll 2
    for (int k0 = 0; k0 < IN_DIM; k0 += 4) {
        // ---- batch of 9 independent loads ----
        v2f a = *(const v2f*)(hrow + k0);
        v2f b1[4], b2[4];
#pragma unroll
        for (int t = 0; t < 4; ++t) {
            b1[t] = *(const v2f*)(w1p + (size_t)t * 16 * IN_DIM + k0);
            b2[t] = *(const v2f*)(w2p + (size_t)t * 16 * IN_DIM + k0);
        }
        // ---- batch of 8 WMMAs ----
#pragma unroll
        for (int t = 0; t < 4; ++t) {
            accz[t] = __builtin_amdgcn_wmma_f32_16x16x4_f32(
                false, a, false, b1[t], (short)0, accz[t], false, false);
            accd[t] = __builtin_amdgcn_wmma_f32_16x16x4_f32(
                false, a, false, b2[t], (short)0, accd[t], false, false);
        }
    }

    // C/D layout: VGPR r -> row = r + 8*(lane>>4), col = lane&15
    const int rplus = (lane >> 4) * 8;
#pragma unroll
    for (int r = 0; r < 8; ++r) {
        const int row = row0 + r + rplus;
        if (row < n) {
#pragma unroll
            for (int t = 0; t < 4; ++t) {
                z [(size_t)row * OUT_DIM + t * 16 + m] = accz[t][r];
                hd[(size_t)row * OUT_DIM + t * 16 + m] = accd[t][r];
            }
        }
    }
}

// ---- Kernel 2: zero agg (in d_out), sums; init segmax keys ----
__global__ void k_init(unsigned* __restrict__ keys, float* __restrict__ sums,
                       float* __restrict__ agg, int n) {
    const int i = blockIdx.x * blockDim.x + threadIdx.x;
    const int total = n * OUT_DIM;
    if (i < total) agg[i] = 0.0f;
    if (i < n) { keys[i] = 0u; sums[i] = 0.0f; }
}

// ---- Kernel 3: edge logits e = leaky_relu((h_d[src]-h_d[dst]) . att), segmax ----
__global__ void __launch_bounds__(256)
k_edge_logit(const float* __restrict__ hd, const int* __restrict__ src,
             const int* __restrict__ dst, const float* __restrict__ att,
             float* __restrict__ e, unsigned* __restrict__ keys, int nedges) {
    const int lane = threadIdx.x & 31;
    const int edge = (int)((blockIdx.x * blockDim.x + threadIdx.x) >> 5);
    if (edge >= nedges) return;
    const int s = src[edge], d = dst[edge];
    const v2f hs = *(const v2f*)(hd + (size_t)s * OUT_DIM + lane * 2);
    const v2f hv = *(const v2f*)(hd + (size_t)d * OUT_DIM + lane * 2);
    const v2f av = *(const v2f*)(att + lane * 2);
    float p = (hs.x - hv.x) * av.x + (hs.y - hv.y) * av.y;
#pragma unroll
    for (int off = 16; off; off >>= 1) p += __shfl_xor(p, off, 32);
    if (lane == 0) {
        const float logit = p > 0.0f ? p : 0.01f * p;  // leaky_relu, slope 0.01
        e[edge] = logit;
        atomicMax(keys + d, f2key(logit));
    }
}

// ---- Kernel 4: ex = exp(e - m[dst]); sums[dst] += ex ----
__global__ void __launch_bounds__(256)
k_edge_exp(const int* __restrict__ dst, float* __restrict__ e,
           const unsigned* __restrict__ keys, float* __restrict__ sums, int nedges) {
    const int i = blockIdx.x * blockDim.x + threadIdx.x;
    if (i >= nedges) return;
    const int d = dst[i];
    const float v = __expf(e[i] - key2f(keys[d]));
    e[i] = v;                 // overwrite own slot with exp value
    atomicAdd(sums + d, v);
}

// ---- Kernel 5: agg[dst] += (ex/sums[dst]) * (h_d[src]-h_d[dst]) ----
__global__ void __launch_bounds__(256)
k_edge_agg(const float* __restrict__ hd, const int* __restrict__ src,
           const int* __restrict__ dst, const float* __restrict__ ex,
           const float* __restrict__ sums, float* __restrict__ agg, int nedges) {
    const int lane = threadIdx.x & 31;
    const int edge = (int)((blockIdx.x * blockDim.x + threadIdx.x) >> 5);
    if (edge >= nedges) return;
    const int s = src[edge], d = dst[edge];
    const float a = ex[edge] / sums[d];
    const v2f hs = *(const v2f*)(hd + (size_t)s * OUT_DIM + lane * 2);
    const v2f hv = *(const v2f*)(hd + (size_t)d * OUT_DIM + lane * 2);
    float* ap = agg + (size_t)d * OUT_DIM + lane * 2;
    atomicAdd(ap,     a * (hs.x - hv.x));
    atomicAdd(ap + 1, a * (hs.y - hv.y));
}

// ---- Kernel 6: out = elu(z + agg)  (agg lives in d_out; in-place) ----
__global__ void k_final(const float* __restrict__ z, float* __restrict__ out, int total) {
    const int i = blockIdx.x * blockDim.x + threadIdx.x;
    if (i >= total) return;
    const float x = z[i] + out[i];
    out[i] = x > 0.0f ? x : (__expf(x) - 1.0f);
}

extern "C" void kernel_launch(void* const* d_in, const int* in_sizes, int n_in,
                              void* d_out, int out_size, void* d_ws, size_t ws_size,
                              hipStream_t stream) {
    const float* h   = (const float*)d_in[0];
    const float* W1  = (const float*)d_in[1];
    const float* W2  = (const float*)d_in[2];
    const float* att = (const float*)d_in[3];
    const int*   src = (const int*)d_in[4];
    const int*   dst = (const int*)d_in[5];
    float* out = (float*)d_out;

    const int n  = in_sizes[0] / IN_DIM;   // 100000
    const int ne = in_sizes[4];            // 1600000

    // Workspace layout
    char* ws = (char*)d_ws;
    float*    z    = (float*)ws;    ws += (size_t)n * OUT_DIM * sizeof(float);
    float*    hd   = (float*)ws;    ws += (size_t)n * OUT_DIM * sizeof(float);
    float*    e    = (float*)ws;    ws += (size_t)ne * sizeof(float);
    float*    sums = (float*)ws;    ws += (size_t)n * sizeof(float);
    unsigned* keys = (unsigned*)ws;

    // 1) projections via f32 WMMA (one wave per 16-node tile, 4 waves/block)
    const int ntiles = (n + 15) / 16;
    k_gemm<<<(ntiles + 3) / 4, 128, 0, stream>>>(h, W1, W2, z, hd, n);

    // 2) init accumulators (agg in d_out) + segmax keys + sums
    const int total = n * OUT_DIM;
    k_init<<<(total + 255) / 256, 256, 0, stream>>>(keys, sums, out, n);

    // 3) edge logits + segment max (wave-per-edge, 8 edges per 256-thread block)
    k_edge_logit<<<(ne + 7) / 8, 256, 0, stream>>>(hd, src, dst, att, e, keys, ne);

    // 4) exp + segment sum (thread-per-edge)
    k_edge_exp<<<(ne + 255) / 256, 256, 0, stream>>>(dst, e, keys, sums, ne);

    // 5) normalized weighted scatter-sum into d_out (wave-per-edge)
    k_edge_agg<<<(ne + 7) / 8, 256, 0, stream>>>(hd, src, dst, e, sums, out, ne);

    // 6) out = elu(z + agg)
    k_final<<<(total + 255) / 256, 256, 0, stream>>>(z, out, total);
}